// GNNPolicy_70566312673591
// MI455X (gfx1250) — compile-verified
//
#include <hip/hip_runtime.h>
#include <math.h>

typedef __attribute__((ext_vector_type(2))) float v2f;
typedef __attribute__((ext_vector_type(8))) float v8f;

#define F_DIM 128
#define H_DIM 128

__device__ __forceinline__ void atomAddF(float* p, float v) {
  // hardware global_atomic_add_f32 (no CAS loop)
  unsafeAtomicAdd(p, v);
}

// ---- degree accumulation: deg[dst] += 1 ------------------------------------
__global__ void k_deg(const int* __restrict__ ei, float* __restrict__ deg, int E) {
  int i = blockIdx.x * blockDim.x + threadIdx.x;
  if (i >= E) return;
  atomAddF(&deg[ei[E + i]], 1.0f);
}

// ---- dinv = (deg + 1)^-0.5 (in place) --------------------------------------
__global__ void k_dinv(float* __restrict__ deg, int N) {
  int i = blockIdx.x * blockDim.x + threadIdx.x;
  if (i >= N) return;
  deg[i] = 1.0f / sqrtf(deg[i] + 1.0f);
}

// ---- Y[N,128] = X[N,128] @ W[128,128] via V_WMMA_F32_16X16X4_F32 -----------
// W staged in LDS per block (64KB). One wave -> 16x128 output strip
// (8 accumulator tiles). Per K-step of 4: one A b64 load + 16 ds loads
// hoisted ahead of a chain of 8 back-to-back WMMAs.
// A frag (16x4 f32): lane L holds row M=L&15; VGPR v holds K = 2*(L>>4)+v.
// B frag (4x16 f32): lane L holds col N=L&15; VGPR v holds K = 2*(L>>4)+v.
// C/D (16x16 f32):   VGPR r, lane L -> row r+8*(L>>4), col L&15.
__global__ void __launch_bounds__(256) k_gemm_wmma(const float* __restrict__ X,
                                                   const float* __restrict__ W,
                                                   float* __restrict__ Y, int nrows) {
  __shared__ float sW[F_DIM * H_DIM];  // 64KB

  // cooperative stage of W into LDS: 16384 floats / 256 threads = 16 float4 each
  {
    int tid = threadIdx.x;
    const float4* Wv4 = (const float4*)W;
    float4* sWv4 = (float4*)sW;
#pragma unroll
    for (int i = 0; i < (F_DIM * H_DIM / 4) / 256; ++i)
      sWv4[tid + i * 256] = Wv4[tid + i * 256];
  }
  __syncthreads();

  int wave = (int)((blockIdx.x * blockDim.x + threadIdx.x) >> 5);
  int lane = threadIdx.x & 31;
  int tiles = (nrows + 15) >> 4;
  if (wave >= tiles) return;               // whole wave exits together (post-barrier)
  int m0   = wave << 4;
  int mrow = m0 + (lane & 15);
  if (mrow >= nrows) mrow = nrows - 1;     // clamp loads; stores guarded below
  int kb   = (lane >> 4) << 1;             // 0 or 2
  int ncol = lane & 15;

  v8f acc[8] = {};
  const float* ap = X + (size_t)mrow * H_DIM;
  const float* bp = sW + ncol;             // per-lane column base in LDS

  for (int k = 0; k < F_DIM; k += 4) {
    v2f a;
    a.x = ap[k + kb];
    a.y = ap[k + kb + 1];
    // hoist all 8 B fragments for this K-step (constant LDS offsets per t)
    v2f b[8];
#pragma unroll
    for (int t = 0; t < 8; ++t) {
      b[t].x = bp[(k + kb) * H_DIM + t * 16];
      b[t].y = bp[(k + kb + 1) * H_DIM + t * 16];
    }
#pragma unroll
    for (int t = 0; t < 8; ++t) {
      acc[t] = __builtin_amdgcn_wmma_f32_16x16x4_f32(
          /*neg_a=*/false, a, /*neg_b=*/false, b[t],
          /*c_mod=*/(short)0, acc[t], /*reuse_a=*/false, /*reuse_b=*/false);
    }
  }

  int rh = (lane >> 4) << 3;               // 0 or 8
#pragma unroll
  for (int t = 0; t < 8; ++t) {
#pragma unroll
    for (int r = 0; r < 8; ++r) {
      int row = m0 + rh + r;
      if (row < nrows) Y[(size_t)row * H_DIM + t * 16 + ncol] = acc[t][r];
    }
  }
}

// ---- per-edge scatter: agg[dst] += h[src] * dinv[src] (one wave per edge) --
__global__ void k_scatter(const int* __restrict__ ei, const float* __restrict__ h,
                          const float* __restrict__ dinv, float* __restrict__ agg, int E) {
  long long gid = (long long)blockIdx.x * blockDim.x + threadIdx.x;
  int e = (int)(gid >> 5), lane = (int)(gid & 31);
  if (e >= E) return;
  int src = ei[e];
  int dst = ei[E + e];
  float s = dinv[src];
  float4 v = ((const float4*)(h + (size_t)src * H_DIM))[lane];
  float* o = agg + (size_t)dst * H_DIM + lane * 4;
  atomAddF(o + 0, v.x * s);
  atomAddF(o + 1, v.y * s);
  atomAddF(o + 2, v.z * s);
  atomAddF(o + 3, v.w * s);
}

// ---- agg = relu(agg*dinv + h*dinv^2 + bias) (in place over agg) ------------
__global__ void k_combine(float* __restrict__ agg, const float* __restrict__ h,
                          const float* __restrict__ dinv, const float* __restrict__ bias,
                          int N) {
  long long i = (long long)blockIdx.x * blockDim.x + threadIdx.x;
  long long total = (long long)N * H_DIM;
  if (i >= total) return;
  int row = (int)(i >> 7), col = (int)(i & 127);
  float di = dinv[row];
  float v = agg[i] * di + h[i] * (di * di) + bias[col];
  agg[i] = fmaxf(v, 0.0f);
}

// ---- segment-sum pooling (one wave per node) -------------------------------
__global__ void k_pool(const float* __restrict__ h, const int* __restrict__ batch,
                       float* __restrict__ pooled, float* __restrict__ counts, int N) {
  long long gid = (long long)blockIdx.x * blockDim.x + threadIdx.x;
  int node = (int)(gid >> 5), lane = (int)(gid & 31);
  if (node >= N) return;
  int g = batch[node];
  float4 v = ((const float4*)(h + (size_t)node * H_DIM))[lane];
  float* p = pooled + (size_t)g * H_DIM + lane * 4;
  atomAddF(p + 0, v.x);
  atomAddF(p + 1, v.y);
  atomAddF(p + 2, v.z);
  atomAddF(p + 3, v.w);
  if (lane == 0) atomAddF(&counts[g], 1.0f);
}

// ---- heads: action_mean[G,A], exp(log_std)[A], value[G] --------------------
__global__ void k_head(const float* __restrict__ pooled, const float* __restrict__ counts,
                       const float* __restrict__ Wa, const float* __restrict__ ba,
                       const float* __restrict__ Wv, const float* __restrict__ bv,
                       const float* __restrict__ lstd, float* __restrict__ out,
                       int G, int A) {
  __shared__ float p[H_DIM];
  int g = blockIdx.x, t = threadIdx.x;
  float invc = 1.0f / fmaxf(counts[g], 1.0f);
  if (t < H_DIM) p[t] = pooled[(size_t)g * H_DIM + t] * invc;
  __syncthreads();
  if (t < A) {
    float s = 0.0f;
    for (int k = 0; k < H_DIM; ++k) s += p[k] * Wa[k * A + t];
    out[g * A + t] = s + ba[t];
    if (g == 0) out[G * A + t] = expf(lstd[t]);
  } else if (t == A) {
    float s = 0.0f;
    for (int k = 0; k < H_DIM; ++k) s += p[k] * Wv[k];
    out[G * A + A + g] = s + bv[0];
  }
}

extern "C" void kernel_launch(void* const* d_in, const int* in_sizes, int n_in,
                              void* d_out, int out_size, void* d_ws, size_t ws_size,
                              hipStream_t stream) {
  const float* x    = (const float*)d_in[0];
  const int*   ei   = (const int*)d_in[1];
  const int*   batch= (const int*)d_in[2];
  const float* W1   = (const float*)d_in[3];
  const float* b1   = (const float*)d_in[4];
  const float* W2   = (const float*)d_in[5];
  const float* b2   = (const float*)d_in[6];
  const float* Wa   = (const float*)d_in[7];
  const float* ba   = (const float*)d_in[8];
  const float* Wv   = (const float*)d_in[9];
  const float* bv   = (const float*)d_in[10];
  const float* lstd = (const float*)d_in[11];
  float* out = (float*)d_out;

  const int N = in_sizes[0] / F_DIM;
  const int E = in_sizes[1] / 2;
  const int A = in_sizes[11];
  const int G = (out_size - A) / (A + 1);

  // workspace carve-up: dinv | bufA (N*H) | bufB (N*H) | pooled (G*H) | counts (G)
  char* w = (char*)d_ws;
  float* dinv = (float*)w;   w += (((size_t)N * 4) + 255) & ~(size_t)255;
  float* bufA = (float*)w;   w += (size_t)N * H_DIM * 4;
  float* bufB = (float*)w;   w += (size_t)N * H_DIM * 4;
  float* pooled = (float*)w; w += (size_t)G * H_DIM * 4;
  float* counts = (float*)w;

  const int tb = 256;
  const unsigned ge  = (unsigned)((E + tb - 1) / tb);
  const unsigned gn  = (unsigned)((N + tb - 1) / tb);
  const unsigned gnh = (unsigned)(((size_t)N * H_DIM + tb - 1) / tb);
  const unsigned gew = (unsigned)(((size_t)E * 32 + tb - 1) / tb);
  const unsigned gnw = (unsigned)(((size_t)N * 32 + tb - 1) / tb);
  const int tiles = (N + 15) / 16;
  const unsigned gg = (unsigned)((tiles + 7) / 8);  // 8 waves / block

  // degree -> dinv
  hipMemsetAsync(dinv, 0, (size_t)N * 4, stream);
  k_deg<<<ge, tb, 0, stream>>>(ei, dinv, E);
  k_dinv<<<gn, tb, 0, stream>>>(dinv, N);

  // layer 1: bufA = x@W1 ; bufB = scatter ; bufB = relu(combine)
  k_gemm_wmma<<<gg, 256, 0, stream>>>(x, W1, bufA, N);
  hipMemsetAsync(bufB, 0, (size_t)N * H_DIM * 4, stream);
  k_scatter<<<gew, tb, 0, stream>>>(ei, bufA, dinv, bufB, E);
  k_combine<<<gnh, tb, 0, stream>>>(bufB, bufA, dinv, b1, N);

  // layer 2: bufA = bufB@W2 ; bufB = scatter ; bufB = relu(combine)
  k_gemm_wmma<<<gg, 256, 0, stream>>>(bufB, W2, bufA, N);
  hipMemsetAsync(bufB, 0, (size_t)N * H_DIM * 4, stream);
  k_scatter<<<gew, tb, 0, stream>>>(ei, bufA, dinv, bufB, E);
  k_combine<<<gnh, tb, 0, stream>>>(bufB, bufA, dinv, b2, N);

  // pool + heads
  hipMemsetAsync(pooled, 0, ((size_t)G * H_DIM + G) * 4, stream);
  k_pool<<<gnw, tb, 0, stream>>>(bufB, batch, pooled, counts, N);
  k_head<<<(unsigned)G, 128, 0, stream>>>(pooled, counts, Wa, ba, Wv, bv, lstd, out, G, A);
}